// ComplexUpSampling2D_18159121728089
// MI455X (gfx1250) — compile-verified
//
#include <hip/hip_runtime.h>
#include <hip/hip_bf16.h>
#include <stdint.h>

// ---------------------------------------------------------------------------
// 2x2 half-pixel bilinear upsample, NHWC f32.
//   in : (16, 128, 128, 128)   out : (16, 256, 256, 128)
// Memory-bound (>= 640 MiB mandatory traffic, ~0.8 GFLOP): the CDNA5 path
// that matters is data movement, not matrix cores:
//   * global_load_async_to_lds_b128 (ASYNCcnt) stages a haloed input tile
//   * ds_load_b128 reads from LDS
//   * non-temporal global_store_b128 (output is write-once/never-read; NT
//     keeps the 128 MiB input resident in the 192 MB L2 so the ~3.4x halo
//     re-reads hit L2 instead of HBM)
// Fixed weights {0.25, 0.75} + clamped indices reproduce the reference's
// edge handling exactly (degenerate bracket -> 0.25v + 0.75v = v).
// ---------------------------------------------------------------------------

typedef float vfloat4 __attribute__((ext_vector_type(4)));

namespace {
constexpr int Bn = 16;
constexpr int H  = 128;
constexpr int Wd = 128;
constexpr int Cn = 128;
constexpr int OH = 2 * H;
constexpr int OW = 2 * Wd;

constexpr int TW_OUT = 32;            // output columns per workgroup tile
constexpr int TM     = TW_OUT / 2;    // 16 "m" centers per tile
constexpr int TW_IN  = TM + 2;        // 18 input columns incl. halo
constexpr int TILE_PIX    = 3 * TW_IN;        // 3 input rows x 18 cols = 54
constexpr int TILE_FLOATS = TILE_PIX * Cn;    // 6912 floats = 27648 B
constexpr int CHUNKS      = TILE_FLOATS / 4;  // 1728 x 16B async chunks
constexpr int BLOCK       = 256;              // 8 wave32
}  // namespace

// Async copy of 16 bytes from global memory at sbase + goff to LDS at lds_dst  (ASYNCcnt path).
// The LDS address operand is derived from the real shared-memory pointer
// (ptrtoint escape) so the compiler must assume the asm writes the tile;
// low 32 bits of a flat LDS pointer are the LDS byte offset (VDST operand).
__device__ __forceinline__ void async_ld_b128(float* lds_dst, uint32_t goff,
                                              const float* sbase) {
  const uint32_t lds_off = (uint32_t)(uintptr_t)lds_dst;
  asm volatile("global_load_async_to_lds_b128 %0, %1, %2"
               :
               : "v"(lds_off), "v"(goff), "s"(sbase)
               : "memory");
}

__device__ __forceinline__ void wait_async_zero() {
#if __has_builtin(__builtin_amdgcn_s_wait_asynccnt)
  __builtin_amdgcn_s_wait_asynccnt(0);
#else
  asm volatile("s_wait_asynccnt 0" ::: "memory");
#endif
}

__global__ __launch_bounds__(BLOCK, 2) void upsample2x2_bilinear(
    const float* __restrict__ in, float* __restrict__ out) {
  __shared__ __align__(16) float tile[TILE_FLOATS];

  const int wt = blockIdx.x;      // 0..7   : W tile
  const int i  = blockIdx.y;      // 0..127 : input row (-> out rows 2i, 2i+1)
  const int b  = blockIdx.z;      // 0..15  : batch
  const int w0 = wt * TM;         // first "m" center of this tile
  const int tid = (int)threadIdx.x;

  const float* inb = in + (size_t)b * (H * Wd * Cn);

  // ---- Stage haloed input tile into LDS via async b128 copies -------------
  // c & 31 : 16B chunk within a pixel's 128 channels (wave lanes cover one
  //          pixel's contiguous 512B -> perfectly coalesced global reads)
  // c >> 5 : tile pixel index (3 rows x 18 cols)
  for (int c = tid; c < CHUNKS; c += BLOCK) {
    const int cc = c & 31;
    const int p  = c >> 5;
    const int lc = p % TW_IN;
    const int lr = p / TW_IN;
    const int ch = cc << 2;
    int gr = i - 1 + lr;  gr = gr < 0 ? 0 : (gr > H - 1 ? H - 1 : gr);
    int gc = w0 - 1 + lc; gc = gc < 0 ? 0 : (gc > Wd - 1 ? Wd - 1 : gc);
    const uint32_t goff = (uint32_t)(((gr * Wd + gc) * Cn + ch) * 4);
    async_ld_b128(&tile[(p << 7) + ch], goff, inb);
  }
  wait_async_zero();
  __syncthreads();

  // ---- Compute: each wave produces one output pixel (128 ch) per iter -----
  const int wave = tid >> 5;            // 0..7 (wave32)
  const int lane = tid & 31;
  const int ch   = lane << 2;           // 4 channels per lane
  float* outb = out + (size_t)b * ((size_t)OH * OW * Cn);

#pragma unroll
  for (int it = 0; it < 8; ++it) {
    const int p2 = it * 8 + wave;       // 0..63 output pixels in tile
    const int ro = p2 >> 5;             // 0: even out row (2i), 1: odd (2i+1)
    const int kx = p2 & 31;             // col within tile
    const int k  = wt * TW_OUT + kx;    // global output column
    const int m  = k >> 1;
    const int lcen = m - w0 + 1;        // local col of "center" input pixel

    int lc0, lc1;  float wc0, wc1;
    if ((k & 1) == 0) { lc0 = lcen - 1; lc1 = lcen;     wc0 = 0.25f; wc1 = 0.75f; }
    else              { lc0 = lcen;     lc1 = lcen + 1; wc0 = 0.75f; wc1 = 0.25f; }

    int lr0, lr1;  float wr0, wr1;
    if (ro == 0)      { lr0 = 0; lr1 = 1; wr0 = 0.25f; wr1 = 0.75f; }
    else              { lr0 = 1; lr1 = 2; wr0 = 0.75f; wr1 = 0.25f; }

    const vfloat4 v00 = *reinterpret_cast<const vfloat4*>(
        &tile[((lr0 * TW_IN + lc0) << 7) + ch]);
    const vfloat4 v01 = *reinterpret_cast<const vfloat4*>(
        &tile[((lr0 * TW_IN + lc1) << 7) + ch]);
    const vfloat4 v10 = *reinterpret_cast<const vfloat4*>(
        &tile[((lr1 * TW_IN + lc0) << 7) + ch]);
    const vfloat4 v11 = *reinterpret_cast<const vfloat4*>(
        &tile[((lr1 * TW_IN + lc1) << 7) + ch]);

    const vfloat4 r = wr0 * (wc0 * v00 + wc1 * v01) +
                      wr1 * (wc0 * v10 + wc1 * v11);

    const int j = 2 * i + ro;
    vfloat4* dst = reinterpret_cast<vfloat4*>(
        &outb[(((size_t)j * OW + k) << 7) + ch]);
#if __has_builtin(__builtin_nontemporal_store)
    __builtin_nontemporal_store(r, dst);   // global_store_b128 th:TH_STORE_NT
#else
    *dst = r;
#endif
  }
}

extern "C" void kernel_launch(void* const* d_in, const int* in_sizes, int n_in,
                              void* d_out, int out_size, void* d_ws,
                              size_t ws_size, hipStream_t stream) {
  (void)in_sizes; (void)n_in; (void)out_size; (void)d_ws; (void)ws_size;
  const float* in = (const float*)d_in[0];
  float* out = (float*)d_out;
  dim3 grid(OW / TW_OUT, H, Bn);   // (8, 128, 16) workgroups
  upsample2x2_bilinear<<<grid, BLOCK, 0, stream>>>(in, out);
}